// PointNet2ClsWithHooks_65171833749933
// MI455X (gfx1250) — compile-verified
//
#include <hip/hip_runtime.h>
#include <hip/hip_bf16.h>

// CDNA5 / gfx1250: wave32, WMMA fp32 16x16x4 path.
typedef float v2f __attribute__((ext_vector_type(2)));
typedef float v8f __attribute__((ext_vector_type(8)));

#define EPSF 1e-5f
#define GW   8     // waves per GEMM block (block = 256 threads)
#define NT   4     // 16-wide o-tiles per wave (64 output columns per block)
#define CH   256   // deterministic BN-stat chunks

// ---------------------------------------------------------------------------
// split x (B,N,6) -> xyz (B,N,3), feats (B,N,3)
__global__ void k_split(const float* __restrict__ x, float* __restrict__ xyz,
                        float* __restrict__ f, long total3) {
  long i = (long)blockIdx.x * blockDim.x + threadIdx.x;
  if (i >= total3) return;
  long p = i / 3; int c = (int)(i % 3);
  xyz[i] = x[p * 6 + c];
  f[i]   = x[p * 6 + 3 + c];
}

// ---------------------------------------------------------------------------
// Farthest point sampling: one workgroup per batch element. Sequential over
// npoint iterations; per-iteration block-wide argmax via LDS tree reduction.
__global__ __launch_bounds__(1024) void k_fps(const float* __restrict__ xyz,
                                              int N, int npoint,
                                              int* __restrict__ out) {
  const int b = blockIdx.x;
  const int t = threadIdx.x;
  const int T = blockDim.x;
  const float* p = xyz + (long)b * N * 3;
  __shared__ float rd[1024];
  __shared__ int   ri[1024];
  __shared__ float cent[3];
  float dloc[4];
#pragma unroll
  for (int j = 0; j < 4; ++j) dloc[j] = 1e10f;
  int cur = 0;
  for (int it = 0; it < npoint; ++it) {
    if (t == 0) {
      out[(long)b * npoint + it] = cur;
      cent[0] = p[cur * 3 + 0];
      cent[1] = p[cur * 3 + 1];
      cent[2] = p[cur * 3 + 2];
    }
    __syncthreads();
    float cx = cent[0], cy = cent[1], cz = cent[2];
    float bd = -1.0f; int bi = 0;
#pragma unroll
    for (int j = 0; j < 4; ++j) {
      int n = t + j * T;
      if (n < N) {
        float dx = p[n * 3 + 0] - cx;
        float dy = p[n * 3 + 1] - cy;
        float dz = p[n * 3 + 2] - cz;
        float dd = dx * dx + dy * dy + dz * dz;
        if (dd < dloc[j]) dloc[j] = dd;
        if (dloc[j] > bd) { bd = dloc[j]; bi = n; }  // ascending n: keeps first max
      }
    }
    rd[t] = bd; ri[t] = bi;
    __syncthreads();
    for (int s = T >> 1; s > 0; s >>= 1) {
      if (t < s) {
        float od = rd[t + s]; int oi = ri[t + s];
        if (od > rd[t] || (od == rd[t] && oi < ri[t])) { rd[t] = od; ri[t] = oi; }
      }
      __syncthreads();
    }
    cur = ri[0];
    __syncthreads();
  }
}

// ---------------------------------------------------------------------------
// gather sampled centroids
__global__ void k_gather(const float* __restrict__ xyz, const int* __restrict__ fps,
                         float* __restrict__ out, int N, int S, int B) {
  int g = blockIdx.x * blockDim.x + threadIdx.x;
  if (g >= B * S) return;
  int b = g / S;
  int n = fps[g];
  const float* src = xyz + ((long)b * N + n) * 3;
  out[(long)g * 3 + 0] = src[0];
  out[(long)g * 3 + 1] = src[1];
  out[(long)g * 3 + 2] = src[2];
}

// ---------------------------------------------------------------------------
// Ball query: first K indices (ascending) within radius; pad with first hit.
__global__ void k_ball(const float* __restrict__ xyz, const float* __restrict__ nxyz,
                       int* __restrict__ idx, int B, int N, int S, int K, float r2) {
  long g = (long)blockIdx.x * blockDim.x + threadIdx.x;
  if (g >= (long)B * S) return;
  int b = (int)(g / S);
  const float* p = xyz + (long)b * N * 3;
  float cx = nxyz[g * 3 + 0], cy = nxyz[g * 3 + 1], cz = nxyz[g * 3 + 2];
  int* o = idx + g * K;
  int cnt = 0, first = 0; bool found = false;
  for (int n = 0; n < N && cnt < K; ++n) {
    float dx = p[n * 3 + 0] - cx;
    float dy = p[n * 3 + 1] - cy;
    float dz = p[n * 3 + 2] - cz;
    if (dx * dx + dy * dy + dz * dz <= r2) {
      if (!found) { found = true; first = n; }
      o[cnt++] = n;
    }
  }
  for (; cnt < K; ++cnt) o[cnt] = first;
}

// ---------------------------------------------------------------------------
// Build grouped matrix G[(b,s,k), 0:3] = xyz[idx]-centroid, G[...,3:] = feats[idx]
__global__ void k_group(const float* __restrict__ xyz, const float* __restrict__ feats,
                        const int* __restrict__ idx, const float* __restrict__ nxyz,
                        float* __restrict__ G, long total, int N, int S, int K, int F) {
  long t = (long)blockIdx.x * blockDim.x + threadIdx.x;
  if (t >= total) return;
  const int C = 3 + F;
  int  c  = (int)(t % C);
  long m  = t / C;
  int  k  = (int)(m % K);
  long bs = m / K;
  int  b  = (int)(bs / S);
  int  n  = idx[bs * K + k];
  float v;
  if (c < 3) v = xyz[((long)b * N + n) * 3 + c] - nxyz[bs * 3 + c];
  else       v = feats[((long)b * N + n) * F + (c - 3)];
  G[t] = v;
}

// SA3 grouped input: concat(l2_xyz(3), l2(256)) -> (rows, 259)
__global__ void k_concat3(const float* __restrict__ xyz, const float* __restrict__ f,
                          float* __restrict__ G, long rows) {
  const int C = 259;
  long t = (long)blockIdx.x * blockDim.x + threadIdx.x;
  if (t >= rows * C) return;
  int c = (int)(t % C); long r = t / C;
  G[t] = (c < 3) ? xyz[r * 3 + c] : f[r * 256 + (c - 3)];
}

// ---------------------------------------------------------------------------
// GEMM  Y[m,o] = sum_c X[m,c]*W[o,c] + bias[o]   via V_WMMA_F32_16X16X4_F32.
// Block = 8 waves x 16 rows = 128 rows; each wave computes NT=4 adjacent
// 16x16 tiles (64 output cols), sharing one A fragment across 4 WMMAs.
// Weight tile (64 rows x Cp) staged in dynamic LDS, zero-padded to Cp.
// Per k-step: load A fragment + all NT B fragments first (clause-able
// ds_load_b64s, one wait), then issue the NT WMMAs back-to-back.
// Fragment layouts per ISA 7.12.2:
//   A (16x4 f32): lane m = lane&15, K pair = (lane>>4)*2.
//   B (4x16 f32): lane n = lane&15, same K pair mapping.
//   D (16x16 f32, 8 VGPRs): vgpr v holds M = v + 8*(lane>>4), N = lane&15.
__global__ __launch_bounds__(256) void k_gemm_wmma(
    const float* __restrict__ X, const float* __restrict__ W,
    const float* __restrict__ bias, float* __restrict__ Y,
    int M, int C, int O) {
  extern __shared__ float wlds[];            // 64 * Cp floats (<=128 KB)
  const int o0 = (int)blockIdx.y * (16 * NT);
  const int Cp = (C + 3) & ~3;
  for (int i = threadIdx.x; i < 64 * Cp; i += blockDim.x) {
    int r = i / Cp, c = i % Cp;
    wlds[i] = (c < C) ? W[(long)(o0 + r) * C + c] : 0.0f;
  }
  __syncthreads();

  const int wave = threadIdx.x >> 5;
  const int lane = threadIdx.x & 31;
  const int m0 = ((int)blockIdx.x * GW + wave) << 4;
  if (m0 >= M) return;                       // wave-uniform exit (EXEC all-1 for WMMA)

  const int nr  = lane & 15;
  const int kof = (lane >> 4) << 1;
  const float* xrow = X + (long)(m0 + nr) * C;
  // per-tile B row bases in LDS; (row*Cp + k0) is 8B-aligned (Cp%4==0, k0 even)
  const float* wr0 = wlds + (long)(0 * 16 + nr) * Cp;
  const float* wr1 = wlds + (long)(1 * 16 + nr) * Cp;
  const float* wr2 = wlds + (long)(2 * 16 + nr) * Cp;
  const float* wr3 = wlds + (long)(3 * 16 + nr) * Cp;

  v8f acc[NT];
#pragma unroll
  for (int t = 0; t < NT; ++t) acc[t] = (v8f){};

  int kb = 0;
  for (; kb + 4 <= C; kb += 4) {             // guard-free main loop
    const int k0 = kb + kof;
    v2f a;
    a.x = xrow[k0];
    a.y = xrow[k0 + 1];
    v2f bf[NT];                              // batch all LDS loads before WMMAs
    bf[0] = *(const v2f*)(wr0 + k0);
    bf[1] = *(const v2f*)(wr1 + k0);
    bf[2] = *(const v2f*)(wr2 + k0);
    bf[3] = *(const v2f*)(wr3 + k0);
#pragma unroll
    for (int t = 0; t < NT; ++t)
      acc[t] = __builtin_amdgcn_wmma_f32_16x16x4_f32(
          false, a, false, bf[t], (short)0, acc[t], false, false);
  }
  if (kb < Cp) {                             // single guarded tail (C%4 != 0)
    const int k0 = kb + kof;
    v2f a;
    a.x = (k0     < C) ? xrow[k0]     : 0.0f;
    a.y = (k0 + 1 < C) ? xrow[k0 + 1] : 0.0f;
    v2f bf[NT];                              // zero-padded in LDS
    bf[0] = *(const v2f*)(wr0 + k0);
    bf[1] = *(const v2f*)(wr1 + k0);
    bf[2] = *(const v2f*)(wr2 + k0);
    bf[3] = *(const v2f*)(wr3 + k0);
#pragma unroll
    for (int t = 0; t < NT; ++t)
      acc[t] = __builtin_amdgcn_wmma_f32_16x16x4_f32(
          false, a, false, bf[t], (short)0, acc[t], false, false);
  }

  const int mb = m0 + ((lane >> 4) << 3);
#pragma unroll
  for (int t = 0; t < NT; ++t) {
    const int oc = o0 + t * 16 + nr;
    const float bv = bias[oc];
#pragma unroll
    for (int v = 0; v < 8; ++v)
      Y[(long)(mb + v) * O + oc] = acc[t][v] + bv;
  }
}

// ---------------------------------------------------------------------------
// Deterministic BN statistics: per-chunk column partial sums, then reduce.
__global__ void k_bnstat_part(const float* __restrict__ Y, long M, int O, long R,
                              float* __restrict__ part) {
  long t = (long)blockIdx.x * blockDim.x + threadIdx.x;
  if (t >= (long)CH * O) return;
  int  o  = (int)(t % O);
  long ch = t / O;
  long r0 = ch * R, r1 = r0 + R; if (r1 > M) r1 = M;
  float s = 0.0f, q = 0.0f;
  for (long m = r0; m < r1; ++m) { float v = Y[m * O + o]; s += v; q += v * v; }
  part[ch * O + o] = s;
  part[(long)CH * O + ch * O + o] = q;
}

__global__ void k_bnfinal(const float* __restrict__ part, const float* __restrict__ g,
                          const float* __restrict__ be, float invM, int O,
                          float* __restrict__ ss) {
  int o = blockIdx.x * blockDim.x + threadIdx.x;
  if (o >= O) return;
  float s = 0.0f, q = 0.0f;
  for (int ch = 0; ch < CH; ++ch) {
    s += part[(long)ch * O + o];
    q += part[(long)CH * O + (long)ch * O + o];
  }
  float mean = s * invM;
  float var  = q * invM - mean * mean;
  float sc = g[o] * rsqrtf(var + EPSF);
  ss[o]     = sc;
  ss[O + o] = be[o] - mean * sc;
}

__global__ void k_bnrelu(float* __restrict__ Y, const float* __restrict__ ss,
                         long total, int O) {
  long i = (long)blockIdx.x * blockDim.x + threadIdx.x;
  long stride = (long)gridDim.x * blockDim.x;
  for (; i < total; i += stride) {
    int o = (int)(i % O);
    Y[i] = fmaxf(0.0f, Y[i] * ss[o] + ss[O + o]);
  }
}

// ---------------------------------------------------------------------------
__global__ void k_maxpool(const float* __restrict__ Y, float* __restrict__ out,
                          long BS, int K, int O) {
  long t = (long)blockIdx.x * blockDim.x + threadIdx.x;
  if (t >= BS * O) return;
  long bs = t / O; int o = (int)(t % O);
  const float* base = Y + bs * (long)K * O + o;
  float v = base[0];
  for (int k = 1; k < K; ++k) v = fmaxf(v, base[(long)k * O]);
  out[t] = v;
}

// tiny classifier head (M=8) — scalar kernels
__global__ void k_fc(const float* __restrict__ X, const float* __restrict__ W,
                     const float* __restrict__ b, float* __restrict__ Y,
                     int M, int C, int O) {
  int t = blockIdx.x * blockDim.x + threadIdx.x;
  if (t >= M * O) return;
  int m = t / O, o = t % O;
  float acc = b[o];
  for (int c = 0; c < C; ++c) acc += X[(long)m * C + c] * W[(long)o * C + c];
  Y[t] = acc;
}

__global__ void k_bn1d_relu(float* __restrict__ Y, const float* __restrict__ g,
                            const float* __restrict__ be, int M, int O) {
  int o = blockIdx.x * blockDim.x + threadIdx.x;
  if (o >= O) return;
  float s = 0.0f, q = 0.0f;
  for (int m = 0; m < M; ++m) { float v = Y[m * O + o]; s += v; q += v * v; }
  float mean = s / M, var = q / M - mean * mean;
  float sc = g[o] * rsqrtf(var + EPSF), sh = be[o] - mean * sc;
  for (int m = 0; m < M; ++m) Y[m * O + o] = fmaxf(0.0f, Y[m * O + o] * sc + sh);
}

// ---------------------------------------------------------------------------
extern "C" void kernel_launch(void* const* d_in, const int* in_sizes, int n_in,
                              void* d_out, int out_size, void* d_ws, size_t ws_size,
                              hipStream_t stream) {
  (void)in_sizes; (void)n_in; (void)out_size; (void)ws_size;
  const int B = 8, N0 = 4096;
  const int S1 = 2048, K1 = 64;   const float R1sq = 0.2f * 0.2f;
  const int S2 = 512,  K2 = 128;  const float R2sq = 0.4f * 0.4f;
  const long M1 = (long)B * S1 * K1;    // 1,048,576
  const long M2 = (long)B * S2 * K2;    //   524,288
  const long M3 = (long)B * 512;        //     4,096

  const float* x = (const float*)d_in[0];
  // input index bases: sa1=1..12, sa2=13..24, sa3=25..36, cls=37..46

  // ---- workspace layout (floats) ----
  float* ws = (float*)d_ws;
  size_t off = 0;
  auto alloc = [&](size_t n) { float* p = ws + off; off += (n + 3) & ~(size_t)3; return p; };
  float* xyz0 = alloc((size_t)B * N0 * 3);
  float* f0   = alloc((size_t)B * N0 * 3);
  int*   fps1 = (int*)alloc((size_t)B * S1);
  float* nx1  = alloc((size_t)B * S1 * 3);
  int*   idx1 = (int*)alloc((size_t)B * S1 * K1);
  float* l1   = alloc((size_t)B * S1 * 128);
  int*   fps2 = (int*)alloc((size_t)B * S2);
  float* nx2  = alloc((size_t)B * S2 * 3);
  int*   idx2 = (int*)alloc((size_t)B * S2 * K2);
  float* l2   = alloc((size_t)B * S2 * 256);
  float* l3   = alloc((size_t)B * 1024);
  float* h1   = alloc((size_t)B * 512);
  float* h2   = alloc((size_t)B * 256);
  float* part = alloc((size_t)2 * CH * 1024);
  float* ss   = alloc((size_t)2 * 1024);
  float* G    = alloc((size_t)M2 * 131);   // max grouped buffer (275 MB)
  float* bufA = alloc((size_t)M1 * 128);   // = M2*256 (537 MB)
  float* bufB = alloc((size_t)M1 * 128);   // ping-pong (537 MB)

  // ---- one MLP layer: WMMA GEMM -> deterministic BN stats -> fused BN+ReLU
  auto mlp_layer = [&](const float* X, int base, int l, float* Y,
                       long M, int C, int O) {
    const float* W  = (const float*)d_in[base + 4 * l + 0];
    const float* bb = (const float*)d_in[base + 4 * l + 1];
    const float* gg = (const float*)d_in[base + 4 * l + 2];
    const float* be = (const float*)d_in[base + 4 * l + 3];
    const int Cp = (C + 3) & ~3;
    dim3 grid((unsigned)((M / 16 + GW - 1) / GW), (unsigned)(O / (16 * NT)));
    size_t shb = (size_t)64 * Cp * sizeof(float);
    k_gemm_wmma<<<grid, 32 * GW, shb, stream>>>(X, W, bb, Y, (int)M, C, O);
    long R  = (M + CH - 1) / CH;
    long nt = (long)CH * O;
    k_bnstat_part<<<(unsigned)((nt + 255) / 256), 256, 0, stream>>>(Y, M, O, R, part);
    k_bnfinal<<<(O + 255) / 256, 256, 0, stream>>>(part, gg, be, 1.0f / (float)M, O, ss);
    k_bnrelu<<<4096, 256, 0, stream>>>(Y, ss, M * (long)O, O);
  };

  // ---- SA1 ----
  long t3 = (long)B * N0 * 3;
  k_split<<<(unsigned)((t3 + 255) / 256), 256, 0, stream>>>(x, xyz0, f0, t3);
  k_fps<<<B, 1024, 0, stream>>>(xyz0, N0, S1, fps1);
  k_gather<<<(B * S1 + 255) / 256, 256, 0, stream>>>(xyz0, fps1, nx1, N0, S1, B);
  k_ball<<<(B * S1 + 255) / 256, 256, 0, stream>>>(xyz0, nx1, idx1, B, N0, S1, K1, R1sq);
  long tg1 = M1 * 6;
  k_group<<<(unsigned)((tg1 + 255) / 256), 256, 0, stream>>>(xyz0, f0, idx1, nx1, G,
                                                             tg1, N0, S1, K1, 3);
  mlp_layer(G,    1, 0, bufA, M1, 6,  64);
  mlp_layer(bufA, 1, 1, bufB, M1, 64, 64);
  mlp_layer(bufB, 1, 2, bufA, M1, 64, 128);
  k_maxpool<<<(unsigned)(((long)B * S1 * 128 + 255) / 256), 256, 0, stream>>>(
      bufA, l1, (long)B * S1, K1, 128);

  // ---- SA2 ----
  k_fps<<<B, 1024, 0, stream>>>(nx1, S1, S2, fps2);
  k_gather<<<(B * S2 + 255) / 256, 256, 0, stream>>>(nx1, fps2, nx2, S1, S2, B);
  k_ball<<<(B * S2 + 255) / 256, 256, 0, stream>>>(nx1, nx2, idx2, B, S1, S2, K2, R2sq);
  long tg2 = M2 * 131;
  k_group<<<(unsigned)((tg2 + 255) / 256), 256, 0, stream>>>(nx1, l1, idx2, nx2, G,
                                                             tg2, S1, S2, K2, 128);
  mlp_layer(G,    13, 0, bufA, M2, 131, 128);
  mlp_layer(bufA, 13, 1, bufB, M2, 128, 128);
  mlp_layer(bufB, 13, 2, bufA, M2, 128, 256);
  k_maxpool<<<(unsigned)(((long)B * S2 * 256 + 255) / 256), 256, 0, stream>>>(
      bufA, l2, (long)B * S2, K2, 256);

  // ---- SA3 (group-all) ----
  long tg3 = M3 * 259;
  k_concat3<<<(unsigned)((tg3 + 255) / 256), 256, 0, stream>>>(nx2, l2, G, M3);
  mlp_layer(G,    25, 0, bufA, M3, 259, 256);
  mlp_layer(bufA, 25, 1, bufB, M3, 256, 512);
  mlp_layer(bufB, 25, 2, bufA, M3, 512, 1024);
  k_maxpool<<<(unsigned)(((long)B * 1024 + 255) / 256), 256, 0, stream>>>(
      bufA, l3, (long)B, 512, 1024);

  // ---- classifier head ----
  const float* W1  = (const float*)d_in[37];
  const float* b1  = (const float*)d_in[38];
  const float* g1  = (const float*)d_in[39];
  const float* be1 = (const float*)d_in[40];
  const float* W2  = (const float*)d_in[41];
  const float* b2  = (const float*)d_in[42];
  const float* g2  = (const float*)d_in[43];
  const float* be2 = (const float*)d_in[44];
  const float* W3  = (const float*)d_in[45];
  const float* b3  = (const float*)d_in[46];
  k_fc<<<(8 * 512 + 255) / 256, 256, 0, stream>>>(l3, W1, b1, h1, 8, 1024, 512);
  k_bn1d_relu<<<2, 256, 0, stream>>>(h1, g1, be1, 8, 512);
  k_fc<<<(8 * 256 + 255) / 256, 256, 0, stream>>>(h1, W2, b2, h2, 8, 512, 256);
  k_bn1d_relu<<<1, 256, 0, stream>>>(h2, g2, be2, 8, 256);
  k_fc<<<1, 32, 0, stream>>>(h2, W3, b3, (float*)d_out, 8, 256, 3);
}